// GenericMoveScorer_39410619908426
// MI455X (gfx1250) — compile-verified
//
#include <hip/hip_runtime.h>
#include <math.h>

#define BB 4
#define NN 20000
#define FF 16
#define EE 320000
#define MM 2048
#define DD 64
#define EPSLN 1e-5f
#define HS 68   // padded LDS row stride (floats)

typedef float v2f __attribute__((ext_vector_type(2)));
typedef float v8f __attribute__((ext_vector_type(8)));

__device__ __forceinline__ v8f wmma4(v2f a, v2f b, v8f c) {
  // D = A(16x4 f32) * B(4x16 f32) + C(16x16 f32)
  return __builtin_amdgcn_wmma_f32_16x16x4_f32(false, a, false, b, (short)0, c,
                                               false, false);
}

__device__ __forceinline__ v8f vzero8() { v8f z = {}; return z; }

__device__ __forceinline__ float silu_f(float x) {
  return x / (1.0f + __expf(-x));
}

// ---------------------------------------------------------------------------
// Time embedding: enc(t)[16] -> 64 (silu) -> 64.   256 threads = 4 batches x 64.
// ---------------------------------------------------------------------------
__global__ void __launch_bounds__(256) k_temb(
    const float* __restrict__ t,
    const float* __restrict__ w1, const float* __restrict__ b1,   // [16][64],[64]
    const float* __restrict__ w2, const float* __restrict__ b2,   // [64][64],[64]
    float* __restrict__ temb)                                     // [4][64]
{
  __shared__ float hid[BB][DD];
  const int d = threadIdx.x & 63;
  const int b = threadIdx.x >> 6;
  const float tv = t[b];
  float enc[16];
#pragma unroll
  for (int j = 0; j < 8; ++j) {
    const float f = __expf(-(float)j * (9.210340371976184f / 8.0f)); // ln(1e4)/8
    const float a = tv * f;
    enc[j] = __sinf(a);
    enc[j + 8] = __cosf(a);
  }
  float s = b1[d];
#pragma unroll
  for (int k = 0; k < 16; ++k) s += enc[k] * w1[k * DD + d];
  hid[b][d] = silu_f(s);
  __syncthreads();
  float o = b2[d];
#pragma unroll
  for (int k = 0; k < DD; ++k) o += hid[b][k] * w2[k * DD + d];
  temb[b * DD + d] = o;
}

// ---------------------------------------------------------------------------
// Node embed: x[80000,16] @ W1[16,64] -> silu -> @ W2[64,64] + b + temb[b]
// One wave = 16 rows x 64 cols via 4 WMMA n-tiles. 8 waves / block.
// ---------------------------------------------------------------------------
__global__ void __launch_bounds__(256) k_embed(
    const float* __restrict__ x,
    const float* __restrict__ w1, const float* __restrict__ b1,
    const float* __restrict__ w2, const float* __restrict__ b2,
    const float* __restrict__ temb,
    float* __restrict__ h)
{
  __shared__ float hid[8][16 * HS];
  const int lane = threadIdx.x & 31, wave = threadIdx.x >> 5;
  const int m16 = lane & 15, half = lane >> 4, dk = half * 2;
  const int row0 = (blockIdx.x * 8 + wave) * 16;     // < 80000, multiple of 16
  const float* arow = x + (size_t)(row0 + m16) * FF;
  float* hw = hid[wave];

  v8f acc[4] = {};
#pragma unroll
  for (int k0 = 0; k0 < FF; k0 += 4) {
    const int c = k0 + dk;
    v2f a = *(const v2f*)(arow + c);
#pragma unroll
    for (int nb = 0; nb < 4; ++nb) {
      const int n = nb * 16 + m16;
      v2f b; b.x = w1[c * DD + n]; b.y = w1[(c + 1) * DD + n];
      acc[nb] = wmma4(a, b, acc[nb]);
    }
  }
#pragma unroll
  for (int nb = 0; nb < 4; ++nb) {
    const int n = nb * 16 + m16;
    const float bias = b1[n];
#pragma unroll
    for (int i = 0; i < 8; ++i)
      hw[(i + half * 8) * HS + n] = silu_f(acc[nb][i] + bias);
  }
  __syncthreads();

#pragma unroll
  for (int nb = 0; nb < 4; ++nb) acc[nb] = vzero8();
  for (int k0 = 0; k0 < DD; k0 += 4) {
    const int c = k0 + dk;
    v2f a = *(const v2f*)(hw + m16 * HS + c);
#pragma unroll
    for (int nb = 0; nb < 4; ++nb) {
      const int n = nb * 16 + m16;
      v2f b; b.x = w2[c * DD + n]; b.y = w2[(c + 1) * DD + n];
      acc[nb] = wmma4(a, b, acc[nb]);
    }
  }
  const int bidx = row0 / NN;                         // tile never crosses batch
#pragma unroll
  for (int nb = 0; nb < 4; ++nb) {
    const int n = nb * 16 + m16;
    const float add = b2[n] + temb[bidx * DD + n];
#pragma unroll
    for (int i = 0; i < 8; ++i)
      h[(size_t)(row0 + i + half * 8) * DD + n] = acc[nb][i] + add;
  }
}

// ---------------------------------------------------------------------------
// Message MLP per edge + scatter-add into agg.
// A tile = 16 edges x 128 (concat h_tgt | h_src), WMMA K=128 -> silu -> K=64.
// ---------------------------------------------------------------------------
__global__ void __launch_bounds__(256) k_msg(
    const float* __restrict__ h,
    const int* __restrict__ ei,                     // [B*E][2] (src,tgt)
    const float* __restrict__ w1, const float* __restrict__ b1,  // [128][64]
    const float* __restrict__ w2, const float* __restrict__ b2,  // [64][64]
    float* __restrict__ agg)
{
  __shared__ float hid[8][16 * HS];
  __shared__ int tgt_s[8][16];
  const int lane = threadIdx.x & 31, wave = threadIdx.x >> 5;
  const int m16 = lane & 15, half = lane >> 4, dk = half * 2;
  const long long e0 = ((long long)blockIdx.x * 8 + wave) * 16;  // flat b*E+e
  const int bidx = (int)(e0 / EE);
  const long long ef = e0 + m16;
  int sn = ei[ef * 2 + 0]; sn = sn < 0 ? 0 : (sn >= NN ? NN - 1 : sn);
  int tn = ei[ef * 2 + 1]; tn = tn < 0 ? 0 : (tn >= NN ? NN - 1 : tn);
  if (lane < 16) tgt_s[wave][m16] = tn;
  const float* pt = h + ((size_t)bidx * NN + tn) * DD;
  const float* ps = h + ((size_t)bidx * NN + sn) * DD;
  float* hw = hid[wave];

  v8f acc[4] = {};
  for (int k0 = 0; k0 < 2 * DD; k0 += 4) {
    const int c = k0 + dk;
    const float* ap = (c < DD) ? (pt + c) : (ps + (c - DD));
    v2f a = *(const v2f*)ap;
#pragma unroll
    for (int nb = 0; nb < 4; ++nb) {
      const int n = nb * 16 + m16;
      v2f b; b.x = w1[c * DD + n]; b.y = w1[(c + 1) * DD + n];
      acc[nb] = wmma4(a, b, acc[nb]);
    }
  }
#pragma unroll
  for (int nb = 0; nb < 4; ++nb) {
    const int n = nb * 16 + m16;
    const float bias = b1[n];
#pragma unroll
    for (int i = 0; i < 8; ++i)
      hw[(i + half * 8) * HS + n] = silu_f(acc[nb][i] + bias);
  }
  __syncthreads();

#pragma unroll
  for (int nb = 0; nb < 4; ++nb) acc[nb] = vzero8();
  for (int k0 = 0; k0 < DD; k0 += 4) {
    const int c = k0 + dk;
    v2f a = *(const v2f*)(hw + m16 * HS + c);
#pragma unroll
    for (int nb = 0; nb < 4; ++nb) {
      const int n = nb * 16 + m16;
      v2f b; b.x = w2[c * DD + n]; b.y = w2[(c + 1) * DD + n];
      acc[nb] = wmma4(a, b, acc[nb]);
    }
  }
#pragma unroll
  for (int nb = 0; nb < 4; ++nb) {
    const int n = nb * 16 + m16;
    const float bias = b2[n];
#pragma unroll
    for (int i = 0; i < 8; ++i) {
      const int node = tgt_s[wave][i + half * 8];
      __hip_atomic_fetch_add(&agg[((size_t)bidx * NN + node) * DD + n],
                             acc[nb][i] + bias,
                             __ATOMIC_RELAXED, __HIP_MEMORY_SCOPE_AGENT);
    }
  }
}

// ---------------------------------------------------------------------------
// Update MLP + residual + layernorm. A tile = 16 nodes x 128 (h | agg).
// Writes h in place (each wave owns its 16 rows).
// ---------------------------------------------------------------------------
__global__ void __launch_bounds__(256) k_upd(
    float* __restrict__ h,
    const float* __restrict__ agg,
    const float* __restrict__ w1, const float* __restrict__ b1,
    const float* __restrict__ w2, const float* __restrict__ b2,
    const float* __restrict__ lng, const float* __restrict__ lnb)
{
  __shared__ float hid[8][16 * HS];
  __shared__ float stat[8][16][2];
  const int lane = threadIdx.x & 31, wave = threadIdx.x >> 5;
  const int m16 = lane & 15, half = lane >> 4, dk = half * 2;
  const int row0 = (blockIdx.x * 8 + wave) * 16;
  const float* ph = h + (size_t)(row0 + m16) * DD;
  const float* pa = agg + (size_t)(row0 + m16) * DD;
  float* hw = hid[wave];

  v8f acc[4] = {};
  for (int k0 = 0; k0 < 2 * DD; k0 += 4) {
    const int c = k0 + dk;
    const float* ap = (c < DD) ? (ph + c) : (pa + (c - DD));
    v2f a = *(const v2f*)ap;
#pragma unroll
    for (int nb = 0; nb < 4; ++nb) {
      const int n = nb * 16 + m16;
      v2f b; b.x = w1[c * DD + n]; b.y = w1[(c + 1) * DD + n];
      acc[nb] = wmma4(a, b, acc[nb]);
    }
  }
#pragma unroll
  for (int nb = 0; nb < 4; ++nb) {
    const int n = nb * 16 + m16;
    const float bias = b1[n];
#pragma unroll
    for (int i = 0; i < 8; ++i)
      hw[(i + half * 8) * HS + n] = silu_f(acc[nb][i] + bias);
  }
  __syncthreads();

#pragma unroll
  for (int nb = 0; nb < 4; ++nb) acc[nb] = vzero8();
  for (int k0 = 0; k0 < DD; k0 += 4) {
    const int c = k0 + dk;
    v2f a = *(const v2f*)(hw + m16 * HS + c);
#pragma unroll
    for (int nb = 0; nb < 4; ++nb) {
      const int n = nb * 16 + m16;
      v2f b; b.x = w2[c * DD + n]; b.y = w2[(c + 1) * DD + n];
      acc[nb] = wmma4(a, b, acc[nb]);
    }
  }
  __syncthreads();
  // x = h + u, overwrite the hidden LDS tile with x
#pragma unroll
  for (int nb = 0; nb < 4; ++nb) {
    const int n = nb * 16 + m16;
    const float bias = b2[n];
#pragma unroll
    for (int i = 0; i < 8; ++i) {
      const int m = i + half * 8;
      hw[m * HS + n] = h[(size_t)(row0 + m) * DD + n] + acc[nb][i] + bias;
    }
  }
  __syncthreads();
  if (lane < 16) {
    float s = 0.f, s2 = 0.f;
    for (int k = 0; k < DD; ++k) {
      const float v = hw[m16 * HS + k];
      s += v; s2 += v * v;
    }
    const float mu = s * (1.0f / DD);
    const float var = s2 * (1.0f / DD) - mu * mu;
    stat[wave][m16][0] = mu;
    stat[wave][m16][1] = rsqrtf(var + EPSLN);
  }
  __syncthreads();
  for (int m = 0; m < 16; ++m) {
    const float mu = stat[wave][m][0], rs = stat[wave][m][1];
#pragma unroll
    for (int j = 0; j < 2; ++j) {
      const int n = lane * 2 + j;
      h[(size_t)(row0 + m) * DD + n] = (hw[m * HS + n] - mu) * rs * lng[n] + lnb[n];
    }
  }
}

// ---------------------------------------------------------------------------
// Scorer: gather 4 node rows per move -> A 16x256 -> 64 (silu) -> 64 (silu) -> 1
// ---------------------------------------------------------------------------
__global__ void __launch_bounds__(256) k_score(
    const float* __restrict__ h,
    const int* __restrict__ mv,                     // [B*M][4]
    const int* __restrict__ mask,                   // [B*M]
    const float* __restrict__ w1, const float* __restrict__ b1,  // [256][64]
    const float* __restrict__ w2, const float* __restrict__ b2,  // [64][64]
    const float* __restrict__ w3, const float* __restrict__ b3,  // [64],[1]
    float* __restrict__ out)
{
  __shared__ float hid[8][16 * HS];
  const int lane = threadIdx.x & 31, wave = threadIdx.x >> 5;
  const int m16 = lane & 15, half = lane >> 4, dk = half * 2;
  const int base = (blockIdx.x * 8 + wave) * 16;    // flat b*M+m
  const int bidx = base / MM;
  const int mvi = base + m16;
  const float* rp[4];
#pragma unroll
  for (int j = 0; j < 4; ++j) {
    int v = mv[mvi * 4 + j];
    v = v < 0 ? 0 : (v >= NN ? NN - 1 : v);
    rp[j] = h + ((size_t)bidx * NN + v) * DD;
  }
  float* hw = hid[wave];

  v8f acc[4] = {};
  for (int k0 = 0; k0 < 4 * DD; k0 += 4) {
    const int c = k0 + dk;
    v2f a = *(const v2f*)(rp[c >> 6] + (c & 63));
#pragma unroll
    for (int nb = 0; nb < 4; ++nb) {
      const int n = nb * 16 + m16;
      v2f b; b.x = w1[c * DD + n]; b.y = w1[(c + 1) * DD + n];
      acc[nb] = wmma4(a, b, acc[nb]);
    }
  }
#pragma unroll
  for (int nb = 0; nb < 4; ++nb) {
    const int n = nb * 16 + m16;
    const float bias = b1[n];
#pragma unroll
    for (int i = 0; i < 8; ++i)
      hw[(i + half * 8) * HS + n] = silu_f(acc[nb][i] + bias);
  }
  __syncthreads();

#pragma unroll
  for (int nb = 0; nb < 4; ++nb) acc[nb] = vzero8();
  for (int k0 = 0; k0 < DD; k0 += 4) {
    const int c = k0 + dk;
    v2f a = *(const v2f*)(hw + m16 * HS + c);
#pragma unroll
    for (int nb = 0; nb < 4; ++nb) {
      const int n = nb * 16 + m16;
      v2f b; b.x = w2[c * DD + n]; b.y = w2[(c + 1) * DD + n];
      acc[nb] = wmma4(a, b, acc[nb]);
    }
  }
  __syncthreads();
#pragma unroll
  for (int nb = 0; nb < 4; ++nb) {
    const int n = nb * 16 + m16;
    const float bias = b2[n];
#pragma unroll
    for (int i = 0; i < 8; ++i)
      hw[(i + half * 8) * HS + n] = silu_f(acc[nb][i] + bias);
  }
  __syncthreads();
  if (lane < 16) {
    float s = b3[0];
    for (int k = 0; k < DD; ++k) s += hw[m16 * HS + k] * w3[k];
    out[mvi] = (mask[mvi] != 0) ? s : -__builtin_inff();
  }
}

// ---------------------------------------------------------------------------
extern "C" void kernel_launch(void* const* d_in, const int* in_sizes, int n_in,
                              void* d_out, int out_size, void* d_ws, size_t ws_size,
                              hipStream_t stream) {
  (void)in_sizes; (void)n_in; (void)out_size; (void)ws_size;
  // data inputs (setup_inputs dict order)
  const float* x    = (const float*)d_in[0];    // [4,20000,16]
  const int*   ei   = (const int*)  d_in[1];    // [4,320000,2]
  const int*   mv   = (const int*)  d_in[2];    // [4,2048,4]
  const int*   mask = (const int*)  d_in[3];    // [4,2048]
  const float* t    = (const float*)d_in[4];    // [4]
  // params: jax tree leaves, dict keys sorted at each level
  // "embed"(5..8), "gnn"(9..48), "scorer"(49..54), "time_embed"(55..58)
  const float* e0b = (const float*)d_in[5];
  const float* e0w = (const float*)d_in[6];
  const float* e1b = (const float*)d_in[7];
  const float* e1w = (const float*)d_in[8];
  struct Layer { const float *lnb,*lng,*m0b,*m0w,*m1b,*m1w,*u0b,*u0w,*u1b,*u1w; } L[4];
  for (int i = 0; i < 4; ++i) {
    const int q = 9 + i * 10;
    L[i].lnb = (const float*)d_in[q + 0];
    L[i].lng = (const float*)d_in[q + 1];
    L[i].m0b = (const float*)d_in[q + 2];
    L[i].m0w = (const float*)d_in[q + 3];
    L[i].m1b = (const float*)d_in[q + 4];
    L[i].m1w = (const float*)d_in[q + 5];
    L[i].u0b = (const float*)d_in[q + 6];
    L[i].u0w = (const float*)d_in[q + 7];
    L[i].u1b = (const float*)d_in[q + 8];
    L[i].u1w = (const float*)d_in[q + 9];
  }
  const float* s0b = (const float*)d_in[49];
  const float* s0w = (const float*)d_in[50];
  const float* s1b = (const float*)d_in[51];
  const float* s1w = (const float*)d_in[52];
  const float* s2b = (const float*)d_in[53];
  const float* s2w = (const float*)d_in[54];
  const float* t0b = (const float*)d_in[55];
  const float* t0w = (const float*)d_in[56];
  const float* t1b = (const float*)d_in[57];
  const float* t1w = (const float*)d_in[58];

  // workspace: h[4*20000*64] | agg[4*20000*64] | temb[4*64]
  float* h    = (float*)d_ws;
  float* agg  = h + (size_t)BB * NN * DD;
  float* temb = agg + (size_t)BB * NN * DD;

  k_temb<<<1, 256, 0, stream>>>(t, t0w, t0b, t1w, t1b, temb);
  k_embed<<<(BB * NN / 16) / 8, 256, 0, stream>>>(x, e0w, e0b, e1w, e1b, temb, h);
  for (int i = 0; i < 4; ++i) {
    hipMemsetAsync(agg, 0, (size_t)BB * NN * DD * sizeof(float), stream);
    k_msg<<<((long long)BB * EE / 16) / 8, 256, 0, stream>>>(
        h, ei, L[i].m0w, L[i].m0b, L[i].m1w, L[i].m1b, agg);
    k_upd<<<(BB * NN / 16) / 8, 256, 0, stream>>>(
        h, agg, L[i].u0w, L[i].u0b, L[i].u1w, L[i].u1b, L[i].lng, L[i].lnb);
  }
  k_score<<<(BB * MM / 16) / 8, 256, 0, stream>>>(
      h, mv, mask, s0w, s0b, s1w, s1b, s2w, s2b, (float*)d_out);
}